// ConditionalHierarchicalCrossEntropyLoss_4475355922683
// MI455X (gfx1250) — compile-verified
//
#include <hip/hip_runtime.h>
#include <math.h>

typedef float v4f __attribute__((ext_vector_type(4)));
typedef float v2f __attribute__((ext_vector_type(2)));
typedef float v8f __attribute__((ext_vector_type(8)));

#define TPB   256        // 8 waves (wave32)
#define NWAVE (TPB / 32)
#define NJ    8          // 8 x float4 x 256 threads = 8192 floats per row
#define MAXD  16

// ---------------------------------------------------------------------------
// Kernel 1: one block per row. Fused row max / sum-exp (softmax stats) over
// y_pred, argmax over y_true, then the <=6-level hierarchical path loss.
// Single pass over both 256MB inputs with non-temporal b128 loads.
// ---------------------------------------------------------------------------
__global__ __launch_bounds__(TPB) void hxent_row_kernel(
    const float* __restrict__ yp, const float* __restrict__ yt,
    const float* __restrict__ cw, const int* __restrict__ paths,
    const int* __restrict__ lens, float* __restrict__ ws,
    int ncls, int Dd)
{
    const int row = blockIdx.x;
    const int t   = threadIdx.x;
    const size_t rowoff = (size_t)row * (size_t)ncls;
    const v4f* yp4 = (const v4f*)(yp + rowoff);
    const v4f* yt4 = (const v4f*)(yt + rowoff);

    // --- stream the row into registers (streaming data: NT hint) -----------
    v4f a[NJ], b[NJ];
#pragma unroll
    for (int j = 0; j < NJ; ++j) a[j] = __builtin_nontemporal_load(&yp4[j * TPB + t]);
#pragma unroll
    for (int j = 0; j < NJ; ++j) b[j] = __builtin_nontemporal_load(&yt4[j * TPB + t]);

    // --- thread-local softmax stats (two-phase in registers: 1 exp/elem) ---
    float mt = -INFINITY;
#pragma unroll
    for (int j = 0; j < NJ; ++j)
        mt = fmaxf(mt, fmaxf(fmaxf(a[j].x, a[j].y), fmaxf(a[j].z, a[j].w)));
    float st = 0.0f;
#pragma unroll
    for (int j = 0; j < NJ; ++j) {
        st += __expf(a[j].x - mt); st += __expf(a[j].y - mt);
        st += __expf(a[j].z - mt); st += __expf(a[j].w - mt);
    }

    // --- thread-local argmax of y_true (first-occurrence semantics) --------
    float bv = -INFINITY; int bi = 0x7fffffff;
#pragma unroll
    for (int j = 0; j < NJ; ++j) {
        const int base = (j * TPB + t) * 4;
        if (b[j].x > bv) { bv = b[j].x; bi = base;     }
        if (b[j].y > bv) { bv = b[j].y; bi = base + 1; }
        if (b[j].z > bv) { bv = b[j].z; bi = base + 2; }
        if (b[j].w > bv) { bv = b[j].w; bi = base + 3; }
    }

    // --- wave32 butterfly reduction ----------------------------------------
#pragma unroll
    for (int off = 16; off > 0; off >>= 1) {
        float mo = __shfl_xor(mt, off);
        float so = __shfl_xor(st, off);
        float mn = fmaxf(mt, mo);
        st = st * __expf(mt - mn) + so * __expf(mo - mn);
        mt = mn;
        float vo = __shfl_xor(bv, off);
        int   io = __shfl_xor(bi, off);
        if (vo > bv || (vo == bv && io < bi)) { bv = vo; bi = io; }
    }

    // --- cross-wave merge via LDS (fixed order => deterministic) -----------
    __shared__ float sm[NWAVE], ss[NWAVE], sv[NWAVE];
    __shared__ int   si[NWAVE];
    const int wave = t >> 5;
    if ((t & 31) == 0) { sm[wave] = mt; ss[wave] = st; sv[wave] = bv; si[wave] = bi; }
    __syncthreads();

    if (t == 0) {
        float M = sm[0], S = ss[0], V = sv[0]; int I = si[0];
#pragma unroll
        for (int w = 1; w < NWAVE; ++w) {
            float mo = sm[w], so = ss[w];
            float mn = fmaxf(M, mo);
            S = S * __expf(M - mn) + so * __expf(mo - mn);
            M = mn;
            if (sv[w] > V || (sv[w] == V && si[w] < I)) { V = sv[w]; I = si[w]; }
        }

        // ---- hierarchical path loss epilogue ------------------------------
        const int   label = I;
        const int   len   = lens[label];
        const float invS  = 1.0f / S;
        const int   Dc    = (Dd < MAXD) ? Dd : MAXD;
        const float* rowp = yp + rowoff;

        float probs[MAXD];
        for (int k = 0; k < Dc; ++k) {
            float pv = 0.0f;
            if (k < len) {
                const int node = paths[label * Dd + k];
                pv = __expf(rowp[node] - M) * invS;   // softmax gathered at node
            }
            probs[k] = pv;
        }
        float suf[MAXD + 1];                          // suffix sums (node+ancestors)
        suf[Dc] = 0.0f;
        for (int k = Dc - 1; k >= 0; --k) suf[k] = suf[k + 1] + probs[k];

        float pl = 0.0f;
        for (int k = 0; k < Dc; ++k) {
            if (k < len - 1) {
                const float cond = suf[k] / (suf[k + 1] + 1e-8f);
                const float h    = (float)(len - 1 - k);
                pl += __expf(-0.1f * h) * __logf(cond + 1e-8f);
            }
        }
        ws[row] = -cw[label] * pl;
    }
}

// ---------------------------------------------------------------------------
// Kernel 2: deterministic mean of B per-row losses. Final 32-lane cross-lane
// sum uses V_WMMA_F32_16X16X4_F32: A = lane values (16x4 f32 layout: lane m
// VGPR0=K0, lane 16+m VGPR0=K2), B = ones => D[m][n] = v[m] + v[16+m].
// lane0 accumulates rowsum[0..7], lane16 rowsum[8..15].
// ---------------------------------------------------------------------------
__global__ __launch_bounds__(TPB) void hxent_reduce_kernel(
    const float* __restrict__ ws, float* __restrict__ out, int B)
{
    __shared__ float part[TPB];
    const int t = threadIdx.x;
    float s = 0.0f;
    for (int i = t; i < B; i += TPB) s += ws[i];
    part[t] = s;
    __syncthreads();

    if (t < 32) {                       // wave 0 fully active => EXEC all ones
        float v = 0.0f;
#pragma unroll
        for (int j = 0; j < TPB / 32; ++j) v += part[t + 32 * j];

        v2f A;  A[0] = v;    A[1] = 0.0f;   // 16x4 A: K1/K3 slots zero
        v2f Bm; Bm[0] = 1.0f; Bm[1] = 1.0f; // B = ones 4x16
        v8f Cm = {};
        Cm = __builtin_amdgcn_wmma_f32_16x16x4_f32(
                 /*neg_a=*/false, A, /*neg_b=*/false, Bm,
                 /*c_mod=*/(short)0, Cm, /*reuse_a=*/false, /*reuse_b=*/false);

        float tsum = Cm[0] + Cm[1] + Cm[2] + Cm[3] +
                     Cm[4] + Cm[5] + Cm[6] + Cm[7];
        float other = __shfl(tsum, 16, 32);  // lane0 <- lane16 partial
        if (t == 0) out[0] = (tsum + other) / (float)B;
    }
}

// ---------------------------------------------------------------------------
extern "C" void kernel_launch(void* const* d_in, const int* in_sizes, int n_in,
                              void* d_out, int out_size, void* d_ws, size_t ws_size,
                              hipStream_t stream) {
    const float* yp    = (const float*)d_in[0];   // y_pred   [B, C] f32
    const float* yt    = (const float*)d_in[1];   // y_true   [B, C] f32
    const float* cwp   = (const float*)d_in[2];   // class_w  [C]    f32
    const int*   paths = (const int*)  d_in[3];   // tree_paths [C, D] i32
    const int*   lensp = (const int*)  d_in[4];   // tree_lens  [C]    i32

    const int C  = in_sizes[2];           // 8192
    const int B  = in_sizes[0] / C;       // 8192
    const int Dd = in_sizes[3] / C;       // 6

    float* ws  = (float*)d_ws;
    float* out = (float*)d_out;

    hxent_row_kernel<<<B, TPB, 0, stream>>>(yp, yt, cwp, paths, lensp, ws, C, Dd);
    hxent_reduce_kernel<<<1, TPB, 0, stream>>>(ws, out, B);
}